// TCN_47330539602315
// MI455X (gfx1250) — compile-verified
//
#include <hip/hip_runtime.h>

// ---------------------------------------------------------------------------
// TCN forecaster for gfx1250 (MI455X), wave32, f32 WMMA 16x16x4.
//
// Design: one workgroup per batch (recurrence is batch-local -> no cross-WG
// sync). 8 waves/WG, each wave owns 8 features. Grouped dilated convs are
// W[32,96] @ im2col[96,Lout] matmuls done with V_WMMA_F32_16X16X4_F32:
//   M = 32 out-channels (2 tiles), N = positions (16-wide tiles),
//   K = 96 = 3 taps x 32 cin (k-major -> each K=4 step stays in one tap).
// B fragments are gathered per-lane from LDS (implicit im2col with dilation).
// Activations use position-stride 80 so the two lane halves (cin vs cin+2,
// 160 dwords apart -> bank offset 32) read disjoint LDS bank sets.
// Weights are pre-packed once into per-lane WMMA A-fragment order in d_ws.
// ---------------------------------------------------------------------------

typedef float v2f __attribute__((ext_vector_type(2)));
typedef float v8f __attribute__((ext_vector_type(8)));

__device__ __forceinline__ v8f wmma4(v2f a, v2f b, v8f c) {
  // D = A(16x4 f32) * B(4x16 f32) + C(16x16 f32)
  return __builtin_amdgcn_wmma_f32_16x16x4_f32(false, a, false, b,
                                               (short)0, c, false, false);
}

// ------------------------- weight pre-pack kernel --------------------------
// pack0  : [f][mt][lane][2]            (layer0, K col 3 zero-padded)
// packMid: [l][f][mt][s(24)][lane][2]  (kappa = 4s + 2*half + j ; k-major:
//                                       k = kappa>>5, cin = kappa&31)
__global__ void tcn_prepack(const float* __restrict__ w0,
                            const float* __restrict__ wmid,
                            float* __restrict__ pack0,
                            float* __restrict__ packMid) {
  int idx = blockIdx.x * blockDim.x + threadIdx.x;
  const int N0 = 64 * 2 * 64;              // 8192
  const int NM = 3 * 64 * 2 * 24 * 64;     // 589824
  if (idx < N0) {
    int j = idx & 1, lane = (idx >> 1) & 31;
    int mt = (idx >> 6) & 1, f = idx >> 7;
    int half = lane >> 4, m = lane & 15;
    int kap = 2 * half + j;                // K column 0..3
    float v = 0.f;
    if (kap < 3) v = w0[(f * 32 + mt * 16 + m) * 3 + kap];
    pack0[idx] = v;
  } else if (idx < N0 + NM) {
    int idx2 = idx - N0;
    int j = idx2 & 1, lane = (idx2 >> 1) & 31;
    int u = idx2 >> 6;
    int s = u % 24; u /= 24;
    int mt = u & 1; u >>= 1;
    int f = u & 63; int l = u >> 6;
    int half = lane >> 4, m = lane & 15;
    int kap = 4 * s + 2 * half + j;        // 0..95
    int k = kap >> 5, cin = kap & 31;
    packMid[idx2] = wmid[((l * 2048 + f * 32 + mt * 16 + m) * 32 + cin) * 3 + k];
  }
}

// ----------------------------- layer helpers -------------------------------
__device__ __forceinline__ void zeroC(v8f* C, int cnt) {
  for (int t = 0; t < cnt; t++) {
#pragma unroll
    for (int i = 0; i < 8; i++) C[t][i] = 0.f;
  }
}

// stores C tiles with bias + relu into LDS buffer (row stride 80)
template <int NT>
__device__ __forceinline__ void storeC(v8f (&C)[2][NT], float* out,
                                       const float* __restrict__ biasPtr,
                                       int n, int half) {
#pragma unroll
  for (int mt = 0; mt < 2; mt++) {
#pragma unroll
    for (int i = 0; i < 8; i++) {
      int row = mt * 16 + 8 * half + i;    // C layout: vgpr i -> rows i / i+8
      float bias = biasPtr[row];
#pragma unroll
      for (int nt = 0; nt < NT; nt++) {
        float v = C[mt][nt][i] + bias;
        out[row * 80 + nt * 16 + n] = v > 0.f ? v : 0.f;
      }
    }
  }
}

// Layer 0: 1 -> 32 channels, K=3 taps (padded to 4), dilation 1, input = e row
__device__ __forceinline__ void layer0(const float* __restrict__ A0,
                                       const float* eRow, float* out,
                                       const float* __restrict__ biasPtr,
                                       int lane, int n, int half) {
  v8f C[2][4];
  zeroC(&C[0][0], 8);
  v2f a0 = *(const v2f*)(A0 + lane * 2);        // mt=0
  v2f a1 = *(const v2f*)(A0 + 64 + lane * 2);   // mt=1
  const float* bsrc = eRow + n + 2 * half;      // B row kappa = 2*half + j
#pragma unroll
  for (int nt = 0; nt < 4; nt++) {
    v2f bf;
    bf.x = bsrc[nt * 16];
    bf.y = bsrc[nt * 16 + 1];
    C[0][nt] = wmma4(a0, bf, C[0][nt]);
    C[1][nt] = wmma4(a1, bf, C[1][nt]);
  }
  storeC<4>(C, out, biasPtr, n, half);
}

// Mid layer: 32 -> 32 channels, K=96 (24 WMMA k-steps), dilation d, relu
template <int NT>
__device__ __forceinline__ void midLayer(const float* __restrict__ Apack,
                                         const float* in, float* out,
                                         const float* __restrict__ biasPtr,
                                         int d, int lane, int n, int half) {
  v8f C[2][NT];
  zeroC(&C[0][0], 2 * NT);
  for (int s = 0; s < 24; s++) {
    v2f a0 = *(const v2f*)(Apack + s * 64 + lane * 2);
    v2f a1 = *(const v2f*)(Apack + 1536 + s * 64 + lane * 2);
    int k = s >> 3;                              // tap index (k-major K order)
    int cin = ((s & 7) << 2) + 2 * half;         // input channel for this lane
    const float* bsrc = in + cin * 80 + k * d + n;
#pragma unroll
    for (int nt = 0; nt < NT; nt++) {
      v2f bf;
      bf.x = bsrc[nt * 16];                      // cin
      bf.y = bsrc[nt * 16 + 80];                 // cin + 1
      C[0][nt] = wmma4(a0, bf, C[0][nt]);
      C[1][nt] = wmma4(a1, bf, C[1][nt]);
    }
  }
  storeC<NT>(C, out, biasPtr, n, half);
}

// ------------------------------- main kernel -------------------------------
__global__ void __launch_bounds__(256) tcn_main(
    const float* __restrict__ x,      // [64,63,64]
    const float* __restrict__ xf,     // [64,12,63]
    const float* __restrict__ b0,     // [2048]
    const float* __restrict__ bmid,   // [3,2048]
    const float* __restrict__ wlast,  // [64,32,3]
    const float* __restrict__ blast,  // [64]
    const float* __restrict__ linW,   // [127]
    const float* __restrict__ linb,   // [1]
    const float* __restrict__ pack0,
    const float* __restrict__ packMid,
    float* __restrict__ out)          // [64,12]
{
  __shared__ float e[64 * 80];           // extended sequences, stride 80
  __shared__ float act[8][2][32 * 80];   // per-wave ping-pong activations
  __shared__ float red[256];

  const int tid  = threadIdx.x;
  const int b    = blockIdx.x;
  const int lane = tid & 31;
  const int w    = tid >> 5;
  const int half = lane >> 4;
  const int n    = lane & 15;

  // zero activations (keeps garbage N-columns finite), build extended input:
  // p<63: x window; 63<=p<74: forecast rows (f<63); f==63 filled per step.
  for (int i = tid; i < 8 * 2 * 32 * 80; i += 256) (&act[0][0][0])[i] = 0.f;
  for (int i = tid; i < 64 * 80; i += 256) {
    int f = i / 80, p = i - f * 80;
    float v = 0.f;
    if (p < 63)                 v = x[(b * 63 + p) * 64 + f];
    else if (p < 74 && f < 63)  v = xf[(b * 12 + (p - 63)) * 63 + f];
    e[i] = v;
  }
  __syncthreads();

  for (int t = 0; t < 12; t++) {
    float waveAcc = 0.f;
    for (int fi = 0; fi < 8; fi++) {
      int f = w * 8 + fi;
      float* buf0 = &act[w][0][0];
      float* buf1 = &act[w][1][0];

      layer0(pack0 + f * 128, &e[f * 80 + t], buf0, b0 + f * 32, lane, n, half);
      midLayer<4>(packMid + (0 * 64 + f) * 3072, buf0, buf1,
                  bmid + 0 * 2048 + f * 32, 2, lane, n, half);
      midLayer<4>(packMid + (1 * 64 + f) * 3072, buf1, buf0,
                  bmid + 1 * 2048 + f * 32, 4, lane, n, half);
      midLayer<3>(packMid + (2 * 64 + f) * 3072, buf0, buf1,
                  bmid + 2 * 2048 + f * 32, 8, lane, n, half);

      // last layer (32 -> 1, dilation 16, no relu): lane = cin partial,
      // fold linW[f] per-lane; blast handled in the scalar reduction.
      float p = 0.f;
#pragma unroll
      for (int k = 0; k < 3; k++)
        p += wlast[(f * 32 + lane) * 3 + k] * buf1[lane * 80 + k * 16];
      waveAcc += p * linW[f];
    }
    red[tid] = waveAcc;
    __syncthreads();
    if (tid == 0) {
      float s = linb[0];
      for (int i = 0; i < 256; i++) s += red[i];                 // conv feats
      for (int j = 0; j < 63; j++)                               // forecast feats
        s += xf[(b * 12 + t) * 63 + j] * linW[64 + j];
      for (int f2 = 0; f2 < 64; f2++) s += blast[f2] * linW[f2]; // last-conv bias
      out[b * 12 + t] = s;
      if (t < 11) e[63 * 80 + 63 + t] = s;   // autoregressive append (f = 63)
    }
    __syncthreads();
  }
}

// ------------------------------- launcher ----------------------------------
extern "C" void kernel_launch(void* const* d_in, const int* in_sizes, int n_in,
                              void* d_out, int out_size, void* d_ws, size_t ws_size,
                              hipStream_t stream) {
  (void)in_sizes; (void)n_in; (void)out_size; (void)ws_size;
  const float* x     = (const float*)d_in[0];
  const float* xf    = (const float*)d_in[1];
  const float* w0    = (const float*)d_in[2];
  const float* b0    = (const float*)d_in[3];
  const float* wmid  = (const float*)d_in[4];
  const float* bmid  = (const float*)d_in[5];
  const float* wlast = (const float*)d_in[6];
  const float* blast = (const float*)d_in[7];
  const float* linW  = (const float*)d_in[8];
  const float* linb  = (const float*)d_in[9];

  float* pack0   = (float*)d_ws;           // 8192 floats
  float* packMid = pack0 + 64 * 2 * 64;    // 589824 floats (total ws ~2.4 MB)

  const int total = 64 * 2 * 64 + 3 * 64 * 2 * 24 * 64;   // 598016
  tcn_prepack<<<(total + 255) / 256, 256, 0, stream>>>(w0, wmid, pack0, packMid);
  tcn_main<<<64, 256, 0, stream>>>(x, xf, b0, bmid, wlast, blast, linW, linb,
                                   pack0, packMid, (float*)d_out);
}